// JosephusGNN_88888643158304
// MI455X (gfx1250) — compile-verified
//
#include <hip/hip_runtime.h>

#define D 128

typedef float v2f __attribute__((ext_vector_type(2)));
typedef float v8f __attribute__((ext_vector_type(8)));

__device__ __forceinline__ void atomAddF(float* p, float v) {
    __hip_atomic_fetch_add(p, v, __ATOMIC_RELAXED, __HIP_MEMORY_SCOPE_AGENT);
}

// ---------------------------------------------------------------- utilities
__global__ void fill_f32(float* __restrict__ p, float v, int n) {
    int i = blockIdx.x * blockDim.x + threadIdx.x;
    if (i < n) p[i] = v;
}

// deg[dst] += 1 for every real edge (self-loop handled by init to 1.0)
__global__ void edge_deg(const long long* __restrict__ ei, float* __restrict__ deg, int E) {
    int e = blockIdx.x * blockDim.x + threadIdx.x;
    if (e < E) {
        long long d = ei[(size_t)E + e];
        atomAddF(&deg[d], 1.0f);
    }
}

// in-place deg -> deg^{-1/2}  (deg >= 1 always, so no zero guard needed)
__global__ void deg_to_dinv(float* __restrict__ deg, int n) {
    int i = blockIdx.x * blockDim.x + threadIdx.x;
    if (i < n) deg[i] = rsqrtf(deg[i]);
}

// ---------------------------------------------------------------- fp32 WMMA GEMM
// C[N x 128] = A[N x 128] * W[128 x 128]
// block = (32, 8): 8 waves, wave y owns column tile y (16 cols); block covers 16 rows.
// V_WMMA_F32_16X16X4_F32 layouts (ISA 7.12.2):
//   A 16x4 : lane L(0-15) V{0,1} = A[M=L, K=2*hi+{0,1}], hi = lane>=16
//   B 4x16 : mirrored, V{0,1} = W[K=2*hi+{0,1}, N=lane&15]
//   C 16x16: VGPR g -> M = g + 8*hi, N = lane&15
__global__ void __launch_bounds__(256)
gemm_wmma_f32(const float* __restrict__ A, const float* __restrict__ W,
              float* __restrict__ C, int nRows) {
    const int lane    = threadIdx.x;      // 0..31
    const int hi      = lane >> 4;        // 0 / 1
    const int l       = lane & 15;
    const int colBase = threadIdx.y * 16; // wave -> 16-col tile
    const int rowBase = blockIdx.x * 16;

    int rowA = rowBase + l;
    if (rowA >= nRows) rowA = nRows - 1;          // clamp (no divergence)
    const float* __restrict__ arow = A + (size_t)rowA * D;

    v8f acc = {};
    for (int k = 0; k < D; k += 4) {
        v2f a, b;
        const int ka = k + 2 * hi;
        a.x = arow[ka + 0];
        a.y = arow[ka + 1];
        b.x = W[(size_t)(ka + 0) * D + colBase + l];
        b.y = W[(size_t)(ka + 1) * D + colBase + l];
        acc = __builtin_amdgcn_wmma_f32_16x16x4_f32(
                  false, a, false, b, (short)0, acc, false, false);
    }

    // epilogue: rows rowBase+8*hi+g, column colBase+l; 8 stores with immediate
    // offsets g*512B off one base. Guard is a single UNIFORM branch (grid is
    // exact for N % 16 == 0, which the launcher guarantees here).
    float* __restrict__ cp = C + (size_t)(rowBase + 8 * hi) * D + colBase + l;
    if (rowBase + 16 <= nRows) {
#pragma unroll
        for (int g = 0; g < 8; ++g) cp[(size_t)g * D] = acc[g];
    } else {
#pragma unroll
        for (int g = 0; g < 8; ++g)
            if (rowBase + 8 * hi + g < nRows) cp[(size_t)g * D] = acc[g];
    }
}

// ---------------------------------------------------------------- edge scatter
// one wave per edge: agg[dst] += h[src] * dinv[src]*dinv[dst]
__global__ void __launch_bounds__(256)
scatter_edges(const long long* __restrict__ ei, const float* __restrict__ h,
              const float* __restrict__ dinv, float* __restrict__ agg,
              int E, int wavesTotal) {
    int gid  = blockIdx.x * blockDim.x + threadIdx.x;
    int wave = gid >> 5;
    int lane = gid & 31;
    for (int e = wave; e < E; e += wavesTotal) {
        long long s = ei[e];
        long long d = ei[(size_t)E + e];
        float nrm = dinv[s] * dinv[d];
        const float4 v = *(const float4*)(h + (size_t)s * D + lane * 4);
        float* o = agg + (size_t)d * D + lane * 4;
        atomAddF(o + 0, v.x * nrm);
        atomAddF(o + 1, v.y * nrm);
        atomAddF(o + 2, v.z * nrm);
        atomAddF(o + 3, v.w * nrm);
    }
}

// out = relu(agg + hlin * dinv^2 (self-loop) + bias)   (out may alias agg or hlin)
__global__ void __launch_bounds__(256)
finalize_relu(const float* __restrict__ agg, const float* __restrict__ hlin,
              const float* __restrict__ dinv, const float* __restrict__ bias,
              float* __restrict__ out, int n) {
    int i = blockIdx.x * blockDim.x + threadIdx.x;
    if (i < n * D) {
        int node = i >> 7;
        int c    = i & (D - 1);
        float di = dinv[node];
        float v  = agg[i] + hlin[i] * di * di + bias[c];
        out[i] = v > 0.0f ? v : 0.0f;
    }
}

// ---------------------------------------------------------------- pooling
// one wave per node: gsum[batch[i]] += h[i]; cnt[batch[i]] += 1
__global__ void __launch_bounds__(256)
pool_scatter(const float* __restrict__ h, const long long* __restrict__ batch,
             float* __restrict__ gsum, float* __restrict__ cnt, int n) {
    int gid  = blockIdx.x * blockDim.x + threadIdx.x;
    int wave = gid >> 5;
    int lane = gid & 31;
    if (wave < n) {
        long long g = batch[wave];
        const float4 v = *(const float4*)(h + (size_t)wave * D + lane * 4);
        float* o = gsum + (size_t)g * D + lane * 4;
        atomAddF(o + 0, v.x);
        atomAddF(o + 1, v.y);
        atomAddF(o + 2, v.z);
        atomAddF(o + 3, v.w);
        if (lane == 0) atomAddF(&cnt[g], 1.0f);
    }
}

// one wave per graph: out[g] = dot(gsum[g]/max(cnt,1), Wf) + bf
__global__ void __launch_bounds__(256)
final_dot(const float* __restrict__ gsum, const float* __restrict__ cnt,
          const float* __restrict__ Wf, const float* __restrict__ bf,
          float* __restrict__ out, int G) {
    int gid  = blockIdx.x * blockDim.x + threadIdx.x;
    int wave = gid >> 5;
    int lane = gid & 31;
    if (wave < G) {
        float c   = cnt[wave];
        float inv = 1.0f / (c > 1.0f ? c : 1.0f);
        float s   = 0.0f;
        for (int j = lane; j < D; j += 32) s += gsum[(size_t)wave * D + j] * Wf[j];
#pragma unroll
        for (int off = 16; off > 0; off >>= 1) s += __shfl_down(s, off, 32);
        if (lane == 0) out[wave] = s * inv + bf[0];
    }
}

// ---------------------------------------------------------------- launch
extern "C" void kernel_launch(void* const* d_in, const int* in_sizes, int n_in,
                              void* d_out, int out_size, void* d_ws, size_t ws_size,
                              hipStream_t stream) {
    const float*     x     = (const float*)d_in[0];
    const long long* ei    = (const long long*)d_in[1];
    const long long* batch = (const long long*)d_in[2];
    const float*     W1    = (const float*)d_in[3];
    const float*     b1    = (const float*)d_in[4];
    const float*     W2    = (const float*)d_in[5];
    const float*     b2    = (const float*)d_in[6];
    const float*     Wf    = (const float*)d_in[7];
    const float*     bf    = (const float*)d_in[8];

    const int N = in_sizes[0] / D;   // 100000
    const int E = in_sizes[1] / 2;   // 1600000
    const int G = out_size;          // 64
    float* out = (float*)d_out;

    // workspace carve-up (256B aligned)
    char* ws = (char*)d_ws;
    size_t off = 0;
    auto carve = [&](size_t bytes) -> float* {
        float* p = (float*)(ws + off);
        off += (bytes + 255) & ~(size_t)255;
        return p;
    };
    float* dinv = carve((size_t)N * sizeof(float));
    float* bufA = carve((size_t)N * D * sizeof(float));
    float* bufB = carve((size_t)N * D * sizeof(float));
    float* gbuf = carve((size_t)(G * D + G) * sizeof(float));
    float* gsum = gbuf;
    float* cnt  = gbuf + (size_t)G * D;
    (void)ws_size; (void)n_in;

    const int T = 256;
    const int ND = N * D;
    const int scatterBlocks = 4096;                 // 32768 persistent waves
    const int wavesTotal = scatterBlocks * (T / 32);
    dim3 gemmGrid((N + 15) / 16), gemmBlock(32, 8);

    // degrees -> dinv (dinv buffer doubles as deg)
    fill_f32<<<(N + T - 1) / T, T, 0, stream>>>(dinv, 1.0f, N);            // self-loop
    edge_deg<<<(E + T - 1) / T, T, 0, stream>>>(ei, dinv, E);
    deg_to_dinv<<<(N + T - 1) / T, T, 0, stream>>>(dinv, N);

    // ---- layer 1: hlin1 = x @ W1 (bufA); agg1 (bufB); h1 -> bufA
    gemm_wmma_f32<<<gemmGrid, gemmBlock, 0, stream>>>(x, W1, bufA, N);
    fill_f32<<<(ND + T - 1) / T, T, 0, stream>>>(bufB, 0.0f, ND);
    scatter_edges<<<scatterBlocks, T, 0, stream>>>(ei, bufA, dinv, bufB, E, wavesTotal);
    finalize_relu<<<(ND + T - 1) / T, T, 0, stream>>>(bufB, bufA, dinv, b1, bufA, N);

    // ---- layer 2: hlin2 = h1 @ W2 (bufB); agg2 (bufA); h2 -> bufA
    gemm_wmma_f32<<<gemmGrid, gemmBlock, 0, stream>>>(bufA, W2, bufB, N);
    fill_f32<<<(ND + T - 1) / T, T, 0, stream>>>(bufA, 0.0f, ND);
    scatter_edges<<<scatterBlocks, T, 0, stream>>>(ei, bufB, dinv, bufA, E, wavesTotal);
    finalize_relu<<<(ND + T - 1) / T, T, 0, stream>>>(bufA, bufB, dinv, b2, bufA, N);

    // ---- mean-pool + final projection
    fill_f32<<<(G * D + G + T - 1) / T, T, 0, stream>>>(gbuf, 0.0f, G * D + G);
    pool_scatter<<<(N * 32 + T - 1) / T, T, 0, stream>>>(bufA, batch, gsum, cnt, N);
    final_dot<<<(G * 32 + T - 1) / T, T, 0, stream>>>(gsum, cnt, Wf, bf, out, G);
}